// EuclidGatingEncoder_12790412607644
// MI455X (gfx1250) — compile-verified
//
#include <hip/hip_runtime.h>
#include <hip/hip_bf16.h>

// ---------------------------------------------------------------------------
// 2-layer GCN (GCNConv w/ self-loops + sym deg^-1/2 norm) + global mean pool
// N=100000 nodes, E=1.6M edges, F=128 features everywhere, G=512 graphs.
//
// Strategy (MI455X):
//  - GEMMs (N x 128 @ 128 x 128) via V_WMMA_F32_16X16X4_F32, exact fp32.
//  - Edge aggregation: wave-per-edge, float4 per lane, fp32 global atomics.
//    h (51 MB) is L2-resident (192 MB L2) so random gather/scatter stays on-chip.
// ---------------------------------------------------------------------------

#define FEAT 128
#define NGRAPH 512

typedef float v2f __attribute__((ext_vector_type(2)));
typedef float v8f __attribute__((ext_vector_type(8)));

// ---------------- degree / norm ----------------

__global__ __launch_bounds__(256) void k_deg_init(float* __restrict__ deg, int n) {
    int i = blockIdx.x * blockDim.x + threadIdx.x;
    if (i < n) deg[i] = 1.0f;                       // self-loop contributes 1
}

__global__ __launch_bounds__(256) void k_deg_acc(const int* __restrict__ dst,
                                                 float* __restrict__ deg, int e) {
    int i = blockIdx.x * blockDim.x + threadIdx.x;
    if (i < e) unsafeAtomicAdd(&deg[dst[i]], 1.0f);
}

__global__ __launch_bounds__(256) void k_rsqrt(float* __restrict__ deg, int n) {
    int i = blockIdx.x * blockDim.x + threadIdx.x;
    if (i < n) deg[i] = __frsqrt_rn(deg[i]);        // deg >= 1 always
}

// ---------------- fp32 WMMA GEMM: H[n,128] = X[n,128] @ W[128,128] ----------

__global__ __launch_bounds__(256) void k_gemm_wmma(const float* __restrict__ X,
                                                   const float* __restrict__ W,
                                                   float* __restrict__ H,
                                                   int nrows) {
    const int lane = threadIdx.x & 31;
    const int wave = threadIdx.x >> 5;
    const int tile = blockIdx.x * 8 + wave;         // 8 waves per block
    const int mt   = tile >> 3;                     // row tile (16 rows)
    const int nt   = tile & 7;                      // col tile (16 cols of 128)
    if (mt * 16 >= nrows) return;                   // uniform per wave

    const int half = lane >> 4;                     // 0: K={0,1}  1: K={2,3}
    const int l    = lane & 15;
    const int m0   = mt * 16;
    const int n0   = nt * 16;

    v8f acc = {0.f, 0.f, 0.f, 0.f, 0.f, 0.f, 0.f, 0.f};
    const float* xrow = X + (size_t)(m0 + l) * FEAT;
    const float* wcol = W + (n0 + l);

#pragma unroll
    for (int k = 0; k < FEAT; k += 4) {
        const int ka = k + 2 * half;
        v2f a, b;
        a.x = xrow[ka];
        a.y = xrow[ka + 1];
        b.x = wcol[(size_t)ka * FEAT];
        b.y = wcol[(size_t)(ka + 1) * FEAT];
        acc = __builtin_amdgcn_wmma_f32_16x16x4_f32(false, a, false, b,
                                                    (short)0, acc, false, false);
    }

    // D layout: lane&15 = column, VGPR r = row r + 8*half
    float* out = H + (size_t)(m0 + 8 * half) * FEAT + n0 + l;
#pragma unroll
    for (int r = 0; r < 8; ++r) out[(size_t)r * FEAT] = acc[r];
}

// ---------------- aggregation ----------------

// out[n,:] = in[n,:] * dis[n]^2   (self-loop term; also initializes out)
__global__ __launch_bounds__(256) void k_selfloop(const float* __restrict__ in,
                                                  const float* __restrict__ dis,
                                                  float* __restrict__ out,
                                                  long long total) {
    long long i = (long long)blockIdx.x * blockDim.x + threadIdx.x;
    if (i < total) {
        float d = dis[i >> 7];                      // i / 128
        out[i] = in[i] * d * d;
    }
}

// wave-per-edge scatter: out[dst,:] += in[src,:] * dis[src]*dis[dst]
__global__ __launch_bounds__(256) void k_edge_agg(const int* __restrict__ src,
                                                  const int* __restrict__ dst,
                                                  const float* __restrict__ dis,
                                                  const float* __restrict__ in,
                                                  float* __restrict__ out, int e) {
    int edge = blockIdx.x * (blockDim.x >> 5) + (threadIdx.x >> 5);
    int lane = threadIdx.x & 31;
    if (edge >= e) return;
    int s = src[edge], d = dst[edge];
    float w = dis[s] * dis[d];
    float4 v = *((const float4*)(in + (size_t)s * FEAT) + lane);
    float* o = out + (size_t)d * FEAT + lane * 4;
    unsafeAtomicAdd(o + 0, v.x * w);
    unsafeAtomicAdd(o + 1, v.y * w);
    unsafeAtomicAdd(o + 2, v.z * w);
    unsafeAtomicAdd(o + 3, v.w * w);
}

__global__ __launch_bounds__(256) void k_bias_relu(float* __restrict__ h,
                                                   const float* __restrict__ b,
                                                   long long total) {
    long long i = (long long)blockIdx.x * blockDim.x + threadIdx.x;
    if (i < total) h[i] = fmaxf(h[i] + b[i & (FEAT - 1)], 0.0f);
}

// ---------------- global mean pool ----------------

__global__ __launch_bounds__(256) void k_pool_init(float* __restrict__ sums,
                                                   float* __restrict__ counts,
                                                   int ntot) {
    int i = blockIdx.x * blockDim.x + threadIdx.x;
    if (i < ntot) sums[i] = 0.0f;
    if (i < NGRAPH) counts[i] = 0.0f;
}

__global__ __launch_bounds__(256) void k_pool_acc(const int* __restrict__ batch,
                                                  const float* __restrict__ h,
                                                  float* __restrict__ sums,
                                                  float* __restrict__ counts, int n) {
    int node = blockIdx.x * (blockDim.x >> 5) + (threadIdx.x >> 5);
    int lane = threadIdx.x & 31;
    if (node >= n) return;
    int g = batch[node];
    if (lane == 0) unsafeAtomicAdd(&counts[g], 1.0f);
    float4 v = *((const float4*)(h + (size_t)node * FEAT) + lane);
    float* o = sums + (size_t)g * FEAT + lane * 4;
    unsafeAtomicAdd(o + 0, v.x);
    unsafeAtomicAdd(o + 1, v.y);
    unsafeAtomicAdd(o + 2, v.z);
    unsafeAtomicAdd(o + 3, v.w);
}

__global__ __launch_bounds__(256) void k_pool_div(float* __restrict__ out,
                                                  const float* __restrict__ counts,
                                                  int ntot) {
    int i = blockIdx.x * blockDim.x + threadIdx.x;
    if (i < ntot) out[i] /= fmaxf(counts[i >> 7], 1.0f);
}

// ---------------------------------------------------------------------------

extern "C" void kernel_launch(void* const* d_in, const int* in_sizes, int n_in,
                              void* d_out, int out_size, void* d_ws, size_t ws_size,
                              hipStream_t stream) {
    const float* x    = (const float*)d_in[0];      // [N,128]
    const int*   ei   = (const int*)d_in[1];        // [2,E]
    const int*   bat  = (const int*)d_in[2];        // [N]
    const float* W1   = (const float*)d_in[3];      // [128,128]
    const float* b1   = (const float*)d_in[4];      // [128]
    const float* W2   = (const float*)d_in[5];      // [128,128]
    const float* b2   = (const float*)d_in[6];      // [128]
    float*       out  = (float*)d_out;              // [G,128]

    const int N = in_sizes[0] / FEAT;
    const int E = in_sizes[1] / 2;
    const int* src = ei;
    const int* dst = ei + E;

    // workspace: dis [N] | bufA [N*128] | bufB [N*128] | counts [G]
    float* dis    = (float*)d_ws;
    float* bufA   = dis + ((N + 255) & ~255);
    float* bufB   = bufA + (size_t)N * FEAT;
    float* counts = bufB + (size_t)N * FEAT;

    const long long total = (long long)N * FEAT;
    const int gElem  = (int)((total + 255) / 256);  // elementwise over N*128
    const int gNode  = (N + 255) / 256;
    const int gEdge  = (E + 255) / 256;
    const int gWaveE = (E + 7) / 8;                 // 8 waves per 256-thread block
    const int gWaveN = (N + 7) / 8;
    const int gGemm  = (N + 15) / 16;               // one 16-row stripe per block (8 waves x 8 col tiles)

    // degrees -> dis = deg^-1/2
    k_deg_init<<<gNode, 256, 0, stream>>>(dis, N);
    k_deg_acc<<<gEdge, 256, 0, stream>>>(dst, dis, E);
    k_rsqrt<<<gNode, 256, 0, stream>>>(dis, N);

    // layer 1: bufA = x @ W1 ; bufB = aggregate(bufA) ; relu(+b1)
    k_gemm_wmma<<<gGemm, 256, 0, stream>>>(x, W1, bufA, N);
    k_selfloop<<<gElem, 256, 0, stream>>>(bufA, dis, bufB, total);
    k_edge_agg<<<gWaveE, 256, 0, stream>>>(src, dst, dis, bufA, bufB, E);
    k_bias_relu<<<gElem, 256, 0, stream>>>(bufB, b1, total);

    // layer 2: bufA = bufB @ W2 ; bufB = aggregate(bufA) ; relu(+b2)
    k_gemm_wmma<<<gGemm, 256, 0, stream>>>(bufB, W2, bufA, N);
    k_selfloop<<<gElem, 256, 0, stream>>>(bufA, dis, bufB, total);
    k_edge_agg<<<gWaveE, 256, 0, stream>>>(src, dst, dis, bufA, bufB, E);
    k_bias_relu<<<gElem, 256, 0, stream>>>(bufB, b2, total);

    // global mean pool
    k_pool_init<<<(NGRAPH * FEAT + 255) / 256, 256, 0, stream>>>(out, counts, out_size);
    k_pool_acc<<<gWaveN, 256, 0, stream>>>(bat, bufB, out, counts, N);
    k_pool_div<<<(out_size + 255) / 256, 256, 0, stream>>>(out, counts, out_size);
}